// Jitter_73220602462337
// MI455X (gfx1250) — compile-verified
//
#include <hip/hip_runtime.h>
#include <stdint.h>

// ---------------------------------------------------------------------------
// Jitter (2nd-order Markov time-jitter gather), MI455X / gfx1250.
//   B=16, I=256, T=8192 -> out (16,256,8190) f32.
// Memory-bound: ~268 MB moved -> ~11.5us at 23.3 TB/s. No matrix math -> no
// WMMA; CDNA5 paths used instead:
//   * TDM tensor_load_to_lds (+ s_wait_tensorcnt) stages gather indices
//   * global_store_async_from_lds_b128 (+ s_wait_asynccnt) writes mindex
//   * global_prefetch_b8 streams x rows
// The serial 8189-step Markov chain is parallelized as a prefix scan over
// composition of 9-state transition maps (nibble-packed u64).
// ---------------------------------------------------------------------------

#define NS        8189      // n_steps = T-3
#define T_DIM     8192
#define B_DIM     16
#define I_DIM     256
#define TOUT      8190      // T-2
#define MSTRIDE   8256      // padded mindex row stride (ints, 16B-multiple)
#define TILE_T    2048      // output columns per gather workgroup
#define STAGE_N   2064      // ints staged to LDS per tile (16B multiple)
#define ICH       4         // i-row chunks per tile (I_DIM/ICH rows per WG)

// Markov logits, float32 (p=0.1): generic row [lp, ls, lp];
// special row (prev1=2,prev2=1): [-inf, log(s/(p+s)), log(p/(p+s))]
#define LPC  (-2.3025851f)     // log(0.1)
#define LSC  (-0.22314355f)    // log(0.8)
#define LB1C (-0.11778304f)    // log(0.8/0.9)
#define LB2C (-2.1972246f)     // log(0.1/0.9)

#define IDMAP 0x876543210ull   // identity map over 9 states (9 nibbles)

typedef uint32_t u32x4 __attribute__((ext_vector_type(4)));
typedef uint32_t u32x8 __attribute__((ext_vector_type(8)));

__device__ __forceinline__ uint32_t rotl32(uint32_t x, int n) {
  return (x << n) | (x >> (32 - n));
}

// JAX Threefry-2x32, 20 rounds (key schedule k0,k1,k2=k0^k1^0x1BD11BDA).
__device__ __forceinline__ void threefry2x32(uint32_t k0, uint32_t k1,
                                             uint32_t x0, uint32_t x1,
                                             uint32_t& y0, uint32_t& y1) {
  const uint32_t k2 = k0 ^ k1 ^ 0x1BD11BDAu;
  x0 += k0; x1 += k1;
  x0 += x1; x1 = rotl32(x1, 13); x1 ^= x0;
  x0 += x1; x1 = rotl32(x1, 15); x1 ^= x0;
  x0 += x1; x1 = rotl32(x1, 26); x1 ^= x0;
  x0 += x1; x1 = rotl32(x1, 6);  x1 ^= x0;
  x0 += k1; x1 += k2 + 1u;
  x0 += x1; x1 = rotl32(x1, 17); x1 ^= x0;
  x0 += x1; x1 = rotl32(x1, 29); x1 ^= x0;
  x0 += x1; x1 = rotl32(x1, 16); x1 ^= x0;
  x0 += x1; x1 = rotl32(x1, 24); x1 ^= x0;
  x0 += k2; x1 += k0 + 2u;
  x0 += x1; x1 = rotl32(x1, 13); x1 ^= x0;
  x0 += x1; x1 = rotl32(x1, 15); x1 ^= x0;
  x0 += x1; x1 = rotl32(x1, 26); x1 ^= x0;
  x0 += x1; x1 = rotl32(x1, 6);  x1 ^= x0;
  x0 += k0; x1 += k1 + 3u;
  x0 += x1; x1 = rotl32(x1, 17); x1 ^= x0;
  x0 += x1; x1 = rotl32(x1, 29); x1 ^= x0;
  x0 += x1; x1 = rotl32(x1, 16); x1 ^= x0;
  x0 += x1; x1 = rotl32(x1, 24); x1 ^= x0;
  x0 += k1; x1 += k2 + 4u;
  x0 += x1; x1 = rotl32(x1, 13); x1 ^= x0;
  x0 += x1; x1 = rotl32(x1, 15); x1 ^= x0;
  x0 += x1; x1 = rotl32(x1, 26); x1 ^= x0;
  x0 += x1; x1 = rotl32(x1, 6);  x1 ^= x0;
  x0 += k2; x1 += k0 + 5u;
  y0 = x0; y1 = x1;
}

// Compose 9-entry maps packed as 9 nibbles: h[i] = f[g[i]] (g applied first).
__device__ __forceinline__ unsigned long long compose9(unsigned long long f,
                                                       unsigned long long g) {
  unsigned long long h = 0ull;
#pragma unroll
  for (int i = 0; i < 9; ++i) {
    uint32_t v = (uint32_t)(g >> (4 * i)) & 15u;
    uint32_t e = (uint32_t)(f >> (4 * v)) & 15u;
    h |= (unsigned long long)e << (4 * i);
  }
  return h;
}

// ---------------------------------------------------------------------------
// Kernel 1: per-batch Markov chain via parallel function-composition scan.
// One block per batch, 512 threads x 16 steps. State st = prev1*3 + prev2.
// Step map: f[st] = c(st)*3 + st/3, c(st) = (st==7 ? aB : aA).
// Result is assembled in LDS and pushed to global via async-from-LDS stores.
// ---------------------------------------------------------------------------
__global__ __launch_bounds__(512) void jitter_scan_kernel(
    const int* __restrict__ seed_ptr, int* __restrict__ mindex) {
  __shared__ unsigned long long sf[512];
  __shared__ __align__(16) int smidx[T_DIM];
  const uint32_t seed = (uint32_t)seed_ptr[0];
  const int b = blockIdx.x;
  const int tid = threadIdx.x;
  const int tbase = tid * 16;

  // batch b uses uniforms idx=3b..3b+2 of the 48 per-step draws; pair layout
  // of threefry counter-mode: pair j -> (out[j], out[j+24]).
  const int pbase = (b < 8) ? 3 * b : 3 * b - 24;
  const bool take_lo = (b < 8);

  uint32_t aApk = 0;                    // 2 bits/step: argmax of generic row
  uint32_t aBpk = 0;                    // 1 bit/step:  (argmax special row)-1
  unsigned long long F = IDMAP;         // local composite over 16 steps

  for (int k = 0; k < 16; ++k) {
    const int t = tbase + k;
    if (t < NS) {
      // step key_t from jax.random.split: bits = threefry(key, iota(2*NS)),
      // counter halves of NS; key_t = (bits[2t], bits[2t+1]).
      uint32_t kw0, kw1;
      {
        uint32_t y0, y1;
        uint32_t c0 = (uint32_t)(2 * t);
        if (c0 < (uint32_t)NS) { threefry2x32(0u, seed, c0, c0 + NS, y0, y1); kw0 = y0; }
        else                   { threefry2x32(0u, seed, c0 - NS, c0, y0, y1); kw0 = y1; }
        uint32_t c1 = (uint32_t)(2 * t + 1);
        if (c1 < (uint32_t)NS) { threefry2x32(0u, seed, c1, c1 + NS, y0, y1); kw1 = y0; }
        else                   { threefry2x32(0u, seed, c1 - NS, c1, y0, y1); kw1 = y1; }
      }
      // 3 Gumbel draws for this batch (of the (16,3) block keyed by key_t)
      float gg[3];
      for (int c3 = 0; c3 < 3; ++c3) {
        uint32_t y0, y1;
        threefry2x32(kw0, kw1, (uint32_t)(pbase + c3),
                     (uint32_t)(pbase + c3 + 24), y0, y1);
        uint32_t bits = take_lo ? y0 : y1;
        float u = __uint_as_float((bits >> 9) | 0x3f800000u) - 1.0f;
        u = fmaxf(u, 1.17549435e-38f);           // jax uniform minval=tiny
        gg[c3] = -logf(-logf(u));                // gumbel
      }
      // argmax, generic row (first-index tie rule like jnp.argmax)
      float v0 = LPC + gg[0], v1 = LSC + gg[1], v2 = LPC + gg[2];
      uint32_t aA = 0; float m = v0;
      if (v1 > m) { aA = 1u; m = v1; }
      if (v2 > m) { aA = 2u; }
      // special row: index 0 has -inf logit -> never chosen
      uint32_t aB = ((LB2C + gg[2]) > (LB1C + gg[1])) ? 2u : 1u;
      aApk |= aA << (2 * k);
      aBpk |= (aB - 1u) << k;
      // fold this step into the local composite
      unsigned long long nf = 0ull;
#pragma unroll
      for (int i = 0; i < 9; ++i) {
        uint32_t v = (uint32_t)(F >> (4 * i)) & 15u;
        uint32_t cc = (v == 7u) ? aB : aA;
        uint32_t e = cc * 3u + ((v * 86u) >> 8);   // (v*86)>>8 == v/3, v<9
        nf |= (unsigned long long)e << (4 * i);
      }
      F = nf;
    }
  }

  // Hillis-Steele inclusive scan of map composition across the block
  sf[tid] = F;
  __syncthreads();
  for (int off = 1; off < 512; off <<= 1) {
    unsigned long long gm = IDMAP;
    if (tid >= off) gm = sf[tid - off];
    __syncthreads();
    if (tid >= off) F = compose9(F, gm);
    sf[tid] = F;
    __syncthreads();
  }
  const unsigned long long P = (tid == 0) ? IDMAP : sf[tid - 1];

  // apply exclusive prefix to initial state (prev2=1, prev1=1) -> st = 4
  uint32_t st = (uint32_t)(P >> 16) & 15u;         // entry 4 (4*4 bit shift)
  for (int k = 0; k < 16; ++k) {
    const int t = tbase + k;
    if (t < NS) {
      uint32_t cc = (st == 7u) ? (((aBpk >> k) & 1u) + 1u)
                               : ((aApk >> (2 * k)) & 3u);
      smidx[t + 2] = (int)cc + t + 1;              // mindex = m + pos - 1
      st = cc * 3u + ((st * 86u) >> 8);
    }
  }
  if (tid == 0) { smidx[0] = 0; smidx[1] = 1; smidx[T_DIM - 1] = T_DIM - 1; }
  __syncthreads();

  // push the whole row to global through the async LDS->memory engine
  int* mrow = mindex + b * MSTRIDE;
  for (int q = tid; q < T_DIM / 4; q += 512) {
    unsigned lds_addr = (unsigned)(uintptr_t)(&smidx[q * 4]);
    unsigned long long gaddr = (unsigned long long)(uintptr_t)(mrow + q * 4);
    asm volatile("global_store_async_from_lds_b128 %0, %1, off"
                 :: "v"(gaddr), "v"(lds_addr) : "memory");
  }
  asm volatile("s_wait_asynccnt 0x0" ::: "memory");
}

// ---------------------------------------------------------------------------
// Kernel 2: gather. One block per (tile, i-chunk, batch). The tile's index
// vector is staged into LDS with one TDM tensor_load_to_lds (issued by wave
// 0; EXEC-independent, TENSORcnt-tracked), then I_DIM/ICH rows are gathered
// with near-contiguous loads and coalesced float2 stores.
// ---------------------------------------------------------------------------
__global__ __launch_bounds__(256) void jitter_gather_kernel(
    const float* __restrict__ x, const int* __restrict__ mindex,
    float* __restrict__ out) {
  __shared__ __align__(16) int lidx[STAGE_N];
  const int b = blockIdx.z;
  const int t0 = blockIdx.x * TILE_T;
  const int ic0 = blockIdx.y * (I_DIM / ICH);
  const int tid = threadIdx.x;

  // --- TDM: stage lidx[j] = mindex[b][t0 + j], j in [0, STAGE_N) -----------
  // D# group0: count=1 | lds_addr | global_addr(57b) | type=2.
  // D# group1: data_size=4B; tensor_dim0=tile_dim0=STAGE_N; dims1/2 = 1;
  //            tensor_dim0_stride=STAGE_N. Row padding (MSTRIDE) keeps the
  //            tail over-read inside the buffer.
  if (tid < 32) {                        // one wave issues the DMA
    const uint64_t ga = (uint64_t)(uintptr_t)(mindex + b * MSTRIDE + t0);
    const uint32_t la = (uint32_t)(uintptr_t)(&lidx[0]);
    u32x4 g0;
    g0[0] = 1u;                                      // count=1 (valid, user)
    g0[1] = la;                                      // lds_addr
    g0[2] = (uint32_t)ga;                            // global_addr[31:0]
    g0[3] = (uint32_t)(ga >> 32) | 0x80000000u;      // addr[56:32] | type=2
    u32x8 g1;
    g1[0] = 0x20000u;                                // data_size=2 (4 bytes)
    g1[1] = ((uint32_t)STAGE_N & 0xFFFFu) << 16;     // tensor_dim0[15:0]
    g1[2] = ((uint32_t)STAGE_N >> 16) | (1u << 16);  // tdim0[31:16]|tdim1 lo=1
    g1[3] = ((uint32_t)STAGE_N & 0xFFFFu) << 16;     // tdim1 hi=0|tile_dim0
    g1[4] = 1u | (1u << 16);                         // tile_dim1=1, tile_dim2=1
    g1[5] = (uint32_t)STAGE_N;                       // tensor_dim0_stride lo
    g1[6] = 0u;                                      // stride hi | tdim1_stride
    g1[7] = 0u;
    asm volatile("tensor_load_to_lds %0, %1" :: "s"(g0), "s"(g1) : "memory");
    __builtin_amdgcn_s_wait_tensorcnt(0);
  }
  __syncthreads();

  int rem = TOUT - t0;                       // valid output columns (even)
  if (rem > TILE_T) rem = TILE_T;
  const size_t xb = (size_t)b * I_DIM * T_DIM;
  const size_t ob = (size_t)b * I_DIM * TOUT;

  for (int i = ic0; i < ic0 + I_DIM / ICH; ++i) {
    const float* xrow = x + xb + (size_t)i * T_DIM;
    float* orow = out + ob + (size_t)i * TOUT + t0;
    if (i + 1 < ic0 + I_DIM / ICH)           // stream next row through caches
      __builtin_prefetch(xrow + T_DIM + t0 + tid * 8, 0, 0);
#pragma unroll
    for (int v = 0; v < TILE_T / 512; ++v) {
      const int j = (v * 256 + tid) * 2;     // out[b][i][t0+j]
      if (j < rem) {
        const int s0 = lidx[j + 1];          // mindex[b][t0+1+j]   (LDS)
        const int s1 = lidx[j + 2];          // mindex[b][t0+2+j]   (LDS)
        float2 r;
        r.x = xrow[s0];
        r.y = xrow[s1];
        *(float2*)(orow + j) = r;            // 8B-aligned coalesced store
      }
    }
  }
}

// ---------------------------------------------------------------------------
extern "C" void kernel_launch(void* const* d_in, const int* in_sizes, int n_in,
                              void* d_out, int out_size, void* d_ws,
                              size_t ws_size, hipStream_t stream) {
  (void)in_sizes; (void)n_in; (void)out_size; (void)ws_size;
  const float* x = (const float*)d_in[0];
  const int* seed = (const int*)d_in[1];
  float* out = (float*)d_out;
  int* mindex = (int*)d_ws;                  // B_DIM * MSTRIDE ints (~528 KB)

  jitter_scan_kernel<<<B_DIM, 512, 0, stream>>>(seed, mindex);
  dim3 grid((TOUT + TILE_T - 1) / TILE_T, ICH, B_DIM);
  jitter_gather_kernel<<<grid, 256, 0, stream>>>(x, mindex, out);
}